// AttnOnlyTransformer_55980603736553
// MI455X (gfx1250) — compile-verified
//
#include <hip/hip_runtime.h>
#include <hip/hip_bf16.h>

typedef float v2f __attribute__((ext_vector_type(2)));
typedef float v8f __attribute__((ext_vector_type(8)));

#define D_MODEL 2048
#define SEQ     2048
#define BATCH   8
#define SCALE   0.022097086912079612f   /* 1/sqrt(2048) */

// V_WMMA_F32_16X16X4_F32: A 16x4 f32 (2 VGPRs), B 4x16 f32 (2 VGPRs), C/D 16x16 f32 (8 VGPRs)
__device__ __forceinline__ v8f wmma4(v2f a, v2f b, v8f c) {
  return __builtin_amdgcn_wmma_f32_16x16x4_f32(false, a, false, b, (short)0, c, false, false);
}

// ---------------------------------------------------------------------------
// Build sinusoidal PE [SEQ, D_MODEL] fp32
// ---------------------------------------------------------------------------
__global__ void __launch_bounds__(256) pe_build(float* __restrict__ PE) {
  int idx = blockIdx.x * 256 + threadIdx.x;   // SEQ*D_MODEL threads
  int s = idx >> 11;
  int d = idx & 2047;
  // div_i = exp(2i * (-ln(10000)/D)),  2i = d & ~1
  float f = __expf((float)(d & ~1) * (-9.210340371976184f / (float)D_MODEL));
  float ang = (float)s * f;
  PE[idx] = (d & 1) ? cosf(ang) : sinf(ang);
}

// ---------------------------------------------------------------------------
// fp32 WMMA GEMM: C[M,N] = A[M,K] * B  (B read as B[k*N+n] or transposed B[n*K+k])
// block = 256 threads (8 waves); wave computes a 16x64 strip of C.
// grid = (N/512, M/16)
// ---------------------------------------------------------------------------
template <int TRANSB>
__global__ void __launch_bounds__(256) gemm_wmma(const float* __restrict__ A,
                                                 const float* __restrict__ B,
                                                 float* __restrict__ C,
                                                 int M, int N, int K) {
  const int wave = threadIdx.x >> 5;
  const int lane = threadIdx.x & 31;
  const int half = lane >> 4;      // lane group 0/1 (K-pair selector)
  const int ml   = lane & 15;      // row (A) / col (B,C) within tile
  const int m0 = blockIdx.y * 16;
  const int n0 = (blockIdx.x * 8 + wave) * 64;

  v8f acc[4] = {};
  const float* arow = A + (size_t)(m0 + ml) * K;

  for (int k0 = 0; k0 < K; k0 += 4) {
    // A 16x4 layout: lanes 0-15 hold K={k0,k0+1}, lanes 16-31 hold K={k0+2,k0+3}
    v2f a = *(const v2f*)(arow + k0 + 2 * half);
#pragma unroll
    for (int t = 0; t < 4; ++t) {
      int n = n0 + t * 16 + ml;
      v2f b;
      if (TRANSB) {
        b = *(const v2f*)(B + (size_t)n * K + k0 + 2 * half);
      } else {
        const float* bp = B + (size_t)(k0 + 2 * half) * N + n;
        b.x = bp[0];
        b.y = bp[N];
      }
      acc[t] = wmma4(a, b, acc[t]);
    }
  }
#pragma unroll
  for (int t = 0; t < 4; ++t) {
    float* crow = C + (size_t)(m0 + 8 * half) * N + n0 + t * 16 + ml;
#pragma unroll
    for (int r = 0; r < 8; ++r) crow[(size_t)r * N] = acc[t][r];
  }
}

// ---------------------------------------------------------------------------
// Per-row softmax stats (online max/sum over causal logits). One wave per row.
// logit[q,k] = (M1[q,k] + PEW[q,tk] + PEWT[k,tq] + W[tq,tk]) * SCALE
// ---------------------------------------------------------------------------
__global__ void __launch_bounds__(256) row_stats(const int* __restrict__ tok,
                                                 const float* __restrict__ W,
                                                 const float* __restrict__ M1,
                                                 const float* __restrict__ PEW,
                                                 const float* __restrict__ PEWT,
                                                 float* __restrict__ rmax,
                                                 float* __restrict__ rinv) {
  const int row  = blockIdx.x * 8 + (threadIdx.x >> 5);  // b*SEQ + q
  const int lane = threadIdx.x & 31;
  const int b = row >> 11;
  const int q = row & 2047;
  const int tq = tok[(b << 11) + q];
  const float* m1row  = M1  + (size_t)q * SEQ;
  const float* pewrow = PEW + (size_t)q * D_MODEL;
  const float* wrow   = W   + (size_t)tq * D_MODEL;

  float m = -3.0e38f, s = 0.f;
  for (int k = lane; k <= q; k += 32) {
    int tk = tok[(b << 11) + k];
    float logit = (m1row[k] + pewrow[tk] + PEWT[(size_t)k * D_MODEL + tq] + wrow[tk]) * SCALE;
    float mn = fmaxf(m, logit);
    s = s * __expf(m - mn) + __expf(logit - mn);
    m = mn;
  }
  for (int off = 16; off > 0; off >>= 1) {
    float m2 = __shfl_xor(m, off, 32);
    float s2 = __shfl_xor(s, off, 32);
    float mn = fmaxf(m, m2);
    s = s * __expf(m - mn) + s2 * __expf(m2 - mn);
    m = mn;
  }
  if (lane == 0) { rmax[row] = m; rinv[row] = 1.0f / s; }
}

// ---------------------------------------------------------------------------
// Fused: out[b, q-tile, :] = softmax(sims) @ PE  (dense part).
// Block = (b, 16-q-rows); 8 waves, wave w owns 256 output columns.
// Each k-tile: block builds 16x16 P tile in LDS, each wave does 4x16 WMMAs.
// grid = (SEQ/16, BATCH)
// ---------------------------------------------------------------------------
__global__ void __launch_bounds__(256) attn_out(const int* __restrict__ tok,
                                                const float* __restrict__ W,
                                                const float* __restrict__ PE,
                                                const float* __restrict__ M1,
                                                const float* __restrict__ PEW,
                                                const float* __restrict__ PEWT,
                                                const float* __restrict__ rmax,
                                                const float* __restrict__ rinv,
                                                float* __restrict__ out) {
  __shared__ float p_lds[256];
  const int b    = blockIdx.y;
  const int qt   = blockIdx.x;
  const int wave = threadIdx.x >> 5;
  const int lane = threadIdx.x & 31;
  const int half = lane >> 4;
  const int ml   = lane & 15;
  const int q0   = qt << 4;
  const int d0   = wave << 8;           // 256 columns per wave

  // This thread's (q_local, k_local) element of the P tile
  const int ql = threadIdx.x >> 4;
  const int kl = threadIdx.x & 15;
  const int q  = q0 + ql;
  const int tq = tok[(b << 11) + q];
  const float m_q = rmax[(b << 11) + q];
  const float i_q = rinv[(b << 11) + q];
  const float* m1row  = M1  + (size_t)q * SEQ;
  const float* pewrow = PEW + (size_t)q * D_MODEL;
  const float* wrow   = W   + (size_t)tq * D_MODEL;

  v8f acc[16] = {};                     // 16 x 256 fp32 accumulator (128 VGPRs)

  const int nkt = qt + 1;               // causal: k-tiles 0..qt
  for (int kt = 0; kt < nkt; ++kt) {
    const int k = (kt << 4) + kl;
    float p = 0.f;
    if (k <= q) {
      int tk = tok[(b << 11) + k];
      float logit = (m1row[k] + pewrow[tk] + PEWT[(size_t)k * D_MODEL + tq] + wrow[tk]) * SCALE;
      p = __expf(logit - m_q) * i_q;
    }
    __syncthreads();                    // previous iteration's reads done
    p_lds[(ql << 4) + kl] = p;
    __syncthreads();

#pragma unroll
    for (int k4 = 0; k4 < 16; k4 += 4) {
      v2f a = *(const v2f*)&p_lds[(ml << 4) + k4 + 2 * half];  // A = P[16x4]
      const int krow = (kt << 4) + k4 + 2 * half;
      const float* pe0 = PE + (size_t)krow * D_MODEL + d0 + ml;
#pragma unroll
      for (int t = 0; t < 16; ++t) {
        v2f bb;
        bb.x = pe0[t << 4];             // PE[krow  ][n]
        bb.y = pe0[D_MODEL + (t << 4)]; // PE[krow+1][n]
        acc[t] = wmma4(a, bb, acc[t]);
      }
    }
  }

#pragma unroll
  for (int t = 0; t < 16; ++t) {
    float* crow = out + ((size_t)b * SEQ + q0 + 8 * half) * D_MODEL + d0 + (t << 4) + ml;
#pragma unroll
    for (int r = 0; r < 8; ++r) crow[(size_t)r * D_MODEL] = acc[t][r];
  }
}

// ---------------------------------------------------------------------------
// Sparse (one-hot) part: out[b,q,tok[b,k]] += attn[b,q,k].
// Deterministic fixed-order LDS partial sums (no float atomics).
// grid = BATCH*SEQ blocks (one per output row), 256 threads.
// ---------------------------------------------------------------------------
#define NPART 4
__global__ void __launch_bounds__(256) scatter_out(const int* __restrict__ tok,
                                                   const float* __restrict__ W,
                                                   const float* __restrict__ M1,
                                                   const float* __restrict__ PEW,
                                                   const float* __restrict__ PEWT,
                                                   const float* __restrict__ rmax,
                                                   const float* __restrict__ rinv,
                                                   float* __restrict__ out) {
  __shared__ float pbuf[SEQ];                 // 8 KB
  __shared__ float sbuf[NPART * D_MODEL];     // 32 KB
  const int row = blockIdx.x;                 // b*SEQ + q
  const int b = row >> 11;
  const int q = row & 2047;
  const int tid = threadIdx.x;
  const int tq = tok[(b << 11) + q];
  const float m_q = rmax[row];
  const float i_q = rinv[row];
  const float* m1row  = M1  + (size_t)q * SEQ;
  const float* pewrow = PEW + (size_t)q * D_MODEL;
  const float* wrow   = W   + (size_t)tq * D_MODEL;

  for (int i = tid; i < NPART * D_MODEL; i += 256) sbuf[i] = 0.f;
  for (int k = tid; k < SEQ; k += 256) {
    float p = 0.f;
    if (k <= q) {
      int tk = tok[(b << 11) + k];
      float logit = (m1row[k] + pewrow[tk] + PEWT[(size_t)k * D_MODEL + tq] + wrow[tk]) * SCALE;
      p = __expf(logit - m_q) * i_q;
    }
    pbuf[k] = p;
  }
  __syncthreads();

  if (tid < NPART) {                          // fixed ascending-k order -> deterministic
    const int kper = SEQ / NPART;             // 512
    const int k0 = tid * kper;
    const int kend = min(k0 + kper, q + 1);
    float* sb = sbuf + tid * D_MODEL;
    for (int k = k0; k < kend; ++k) {
      int tk = tok[(b << 11) + k];
      sb[tk] += pbuf[k];
    }
  }
  __syncthreads();

  float* orow = out + (size_t)row * D_MODEL;
  for (int d = tid; d < D_MODEL; d += 256) {
    float s = sbuf[d];
#pragma unroll
    for (int t = 1; t < NPART; ++t) s += sbuf[t * D_MODEL + d];
    orow[d] += s;
  }
}

// ---------------------------------------------------------------------------
extern "C" void kernel_launch(void* const* d_in, const int* in_sizes, int n_in,
                              void* d_out, int out_size, void* d_ws, size_t ws_size,
                              hipStream_t stream) {
  (void)in_sizes; (void)n_in; (void)out_size; (void)ws_size;
  const int*   tok = (const int*)d_in[0];     // [8, 2048] int32
  const float* W   = (const float*)d_in[1];   // [2048, 2048] fp32
  float* out = (float*)d_out;                 // [8, 2048, 2048] fp32

  // Workspace layout (fp32): PE | PEW | PEWT | M1 | rmax | rinv  (~64.1 MB)
  float* ws   = (float*)d_ws;
  float* PE   = ws;
  float* PEW  = ws + (size_t)4 * 1024 * 1024;
  float* PEWT = ws + (size_t)8 * 1024 * 1024;
  float* M1   = ws + (size_t)12 * 1024 * 1024;
  float* rmax = ws + (size_t)16 * 1024 * 1024;
  float* rinv = rmax + BATCH * SEQ;

  pe_build<<<(SEQ * D_MODEL) / 256, 256, 0, stream>>>(PE);
  // PEW  = PE @ W
  gemm_wmma<0><<<dim3(D_MODEL / 512, SEQ / 16), 256, 0, stream>>>(PE, W, PEW, SEQ, D_MODEL, D_MODEL);
  // PEWT = PE @ W^T
  gemm_wmma<1><<<dim3(D_MODEL / 512, SEQ / 16), 256, 0, stream>>>(PE, W, PEWT, SEQ, D_MODEL, D_MODEL);
  // M1   = PEW @ PE^T
  gemm_wmma<1><<<dim3(SEQ / 512, SEQ / 16), 256, 0, stream>>>(PEW, PE, M1, SEQ, SEQ, D_MODEL);
  row_stats<<<(BATCH * SEQ) / 8, 256, 0, stream>>>(tok, W, M1, PEW, PEWT, rmax, rinv);
  attn_out<<<dim3(SEQ / 16, BATCH), 256, 0, stream>>>(tok, W, PE, M1, PEW, PEWT, rmax, rinv, out);
  scatter_out<<<BATCH * SEQ, 256, 0, stream>>>(tok, W, M1, PEW, PEWT, rmax, rinv, out);
}